// ImageSampling_62689342653159
// MI455X (gfx1250) — compile-verified
//
#include <hip/hip_runtime.h>

typedef float v2f __attribute__((ext_vector_type(2)));
typedef float v8f __attribute__((ext_vector_type(8)));

#define BB   4
#define HH   512
#define WW   512
#define CC   64
#define NVV  50000
#define GROUPS_PER_B (NVV / 16)        // 3125 sample-groups of 16 per batch
#define WAVES_PER_BLK 4

// One wave32 handles 16 samples of one batch image.
// Lanes 0-15 own texels (y0,x0)/(y1,x0) and weights (wa,wb) of sample s0+(lane&15);
// lanes 16-31 own texels (y0,x1)/(y1,x1) and weights (wc,wd) of the same sample.
// For each sample t we broadcast its two row offsets across each 16-lane half
// (__shfl width 16 -> ds_bpermute), load the 4x16 f32 B tile per channel block,
// and accumulate with V_WMMA_F32_16X16X4_F32 where A has sample-t's weights in
// row t only. After 16 WMMAs per channel block, D row m holds sample s0+m.
__global__ __launch_bounds__(WAVES_PER_BLK * 32)
void ImageSampling_bilinear_wmma(const float* __restrict__ img,
                                 const float* __restrict__ uv,
                                 float* __restrict__ out) {
    const int lane   = threadIdx.x & 31;
    const int wave   = threadIdx.x >> 5;
    const int wid    = blockIdx.x * WAVES_PER_BLK + wave;   // 0 .. 12499
    const int b      = wid / GROUPS_PER_B;
    const int g      = wid % GROUPS_PER_B;
    const int s0     = g * 16;
    const int lanelo = lane & 15;
    const bool hihalf = (lane >= 16);

    // Every lane loads the uv of "its" sample (both halves load the same 16 uvs -> L2 hit).
    const float2 u = ((const float2*)uv)[(size_t)b * NVV + s0 + lanelo];
    const float x = u.x, y = u.y;
    int x0 = min(max((int)floorf(x), 0), WW - 1);
    int x1 = min(x0 + 1, WW - 1);
    int y0 = min(max((int)floorf(y), 0), HH - 1);
    int y1 = min(y0 + 1, HH - 1);
    const float dx1 = (float)x1 - x, dx0 = x - (float)x0;
    const float dy1 = (float)y1 - y, dy0 = y - (float)y0;

    // k-slot weights held by this lane: lo half -> (wa, wb); hi half -> (wc, wd)
    const float w0 = hihalf ? dx0 * dy1 : dx1 * dy1;
    const float w1 = hihalf ? dx0 * dy0 : dx1 * dy0;
    const int   xs = hihalf ? x1 : x0;

    // byte offsets of the two texel rows this lane is responsible for
    const int off0 = (((b * HH + y0) * WW + xs) * CC) << 2;   // < 2^28, fits i32
    const int off1 = (((b * HH + y1) * WW + xs) * CC) << 2;

    const char* imgb = (const char*)img;

    v8f acc[4] = {{}, {}, {}, {}};   // 16 samples x 64 channels (4 blocks of 16)

#pragma unroll
    for (int t = 0; t < 16; ++t) {
        // broadcast sample t's row offsets within each 16-lane half
        const int o0 = __shfl(off0, t, 16);
        const int o1 = __shfl(off1, t, 16);

        // A matrix: row t (and t+16 for k=2/3 slots) carries sample t's weights
        v2f A;
        A.x = (lanelo == t) ? w0 : 0.0f;   // k0 (lanes 0-15) / k2 (lanes 16-31)
        A.y = (lanelo == t) ? w1 : 0.0f;   // k1 / k3

        const float* p0 = (const float*)(imgb + o0) + lanelo;  // rows k0 / k2
        const float* p1 = (const float*)(imgb + o1) + lanelo;  // rows k1 / k3

#pragma unroll
        for (int cb = 0; cb < 4; ++cb) {
            v2f Bm;
            Bm.x = p0[cb * 16];   // 64B coalesced per half-wave
            Bm.y = p1[cb * 16];
            acc[cb] = __builtin_amdgcn_wmma_f32_16x16x4_f32(
                false, A, false, Bm, (short)0, acc[cb], false, false);
        }
    }

    // D layout: VGPR r -> (sample s0+r, lanes 0-15) and (sample s0+r+8, lanes 16-31)
    float* ob = out + ((size_t)b * NVV + s0 + (hihalf ? 8 : 0)) * CC + lanelo;
#pragma unroll
    for (int cb = 0; cb < 4; ++cb) {
#pragma unroll
        for (int r = 0; r < 8; ++r) {
            __builtin_nontemporal_store(acc[cb][r], ob + (size_t)r * CC + cb * 16);
        }
    }
}

extern "C" void kernel_launch(void* const* d_in, const int* in_sizes, int n_in,
                              void* d_out, int out_size, void* d_ws, size_t ws_size,
                              hipStream_t stream) {
    const float* img = (const float*)d_in[0];   // (B,H,W,C) f32
    const float* uv  = (const float*)d_in[1];   // (B,NV,2) f32
    float* out = (float*)d_out;                 // (B,NV,C) f32

    const int total_waves = BB * GROUPS_PER_B;          // 12500
    dim3 grid(total_waves / WAVES_PER_BLK);             // 3125
    dim3 block(WAVES_PER_BLK * 32);                     // 128 (4 wave32s)
    hipLaunchKernelGGL(ImageSampling_bilinear_wmma, grid, block, 0, stream,
                       img, uv, out);
}